// WeightedAverageWirelength_24816321037008
// MI455X (gfx1250) — compile-verified
//
#include <hip/hip_runtime.h>
#include <stdint.h>
#include <limits.h>

// ---------------------------------------------------------------------------
// WeightedAverageWirelength for MI455X (gfx1250).
// Bandwidth/atomic bound (~350MB HBM/call -> ~15us floor @ 23.3 TB/s).
// Scratch (120MB) is L2-resident (192MB L2), so scattered atomics stay on-chip.
// Heavy pass stages pos (2D tile: rows x,y) + pin2net via the Tensor Data
// Mover into LDS (TENSORcnt), keeping the vector pipe free for gathers+atomics.
// ---------------------------------------------------------------------------

#define THREADS 256
#define CHUNK   2048   // pins staged per block via TDM (tile_dim0 <= 65535)

typedef unsigned int u32x4 __attribute__((ext_vector_type(4)));
typedef int          i32x8 __attribute__((ext_vector_type(8)));
typedef int          i32x4 __attribute__((ext_vector_type(4)));

// Order-preserving float<->int map so integer atomicMin/Max implement
// float min/max (global_atomic_min_i32 / global_atomic_max_i32).
__device__ __forceinline__ int f2ord(float f) {
  int i = __float_as_int(f);
  return (i >= 0) ? i : (i ^ 0x7fffffff);
}
__device__ __forceinline__ float ord2f(int o) {
  return __int_as_float((o >= 0) ? o : (o ^ 0x7fffffff));
}

// ---------------------------------------------------------------------------
// Tensor Data Mover: load tile1 rows of tile0 4-byte elements (row stride
// stride0 elements) from gptr into LDS at byte offset lds_off. Rows land
// back-to-back in LDS. OOB rows/cols (beyond tdim0/tdim1) read as zero.
// D# layout per CDNA5 ISA 8.3/8.4; clang-23 6-arg builtin form.
// ---------------------------------------------------------------------------
__device__ __forceinline__ void tdm_load_4b(uint32_t lds_off, const void* gptr,
                                            uint32_t tile0, uint32_t tile1,
                                            uint32_t tdim0, uint32_t tdim1,
                                            uint32_t stride0) {
  uint64_t ga = (uint64_t)(uintptr_t)gptr;
  u32x4 g0;
  g0.x = 1u;                                                // count=1 (valid)
  g0.y = lds_off;                                           // lds_addr (bytes)
  g0.z = (uint32_t)ga;                                      // global_addr[31:0]
  g0.w = (uint32_t)((ga >> 32) & 0x01ffffffu) | (2u << 30); // addr[56:32], type=2
  i32x8 g1;
  g1.s0 = (int)(2u << 16);                                  // data_size=4B
  g1.s1 = (int)((tdim0 & 0xffffu) << 16);                   // tensor_dim0[15:0] @ [63:48]
  g1.s2 = (int)((tdim0 >> 16) | ((tdim1 & 0xffffu) << 16)); // td0[31:16], td1[15:0]
  g1.s3 = (int)((tdim1 >> 16) | (tile0 << 16));             // td1[31:16], tile_dim0
  g1.s4 = (int)(tile1 & 0xffffu);                           // tile_dim1 (tile_dim2=0)
  g1.s5 = (int)stride0;                                     // tensor_dim0_stride[31:0]
  g1.s6 = 0;                                                // stride hi, td1_stride lo
  g1.s7 = 0;
  i32x4 z4 = {0, 0, 0, 0};
  i32x8 z8 = {0, 0, 0, 0, 0, 0, 0, 0};
  // clang-23 lane: 6 args (g0, g1, g2, g3, extra group, cpol)
  __builtin_amdgcn_tensor_load_to_lds(g0, g1, z4, z4, z8, 0);
}

// ---------------------------------------------------------------------------
// K0: init scratch + output (must rerun every call: graph replay, no repoison)
//   minmax[net] = int4{ordmin_x=INT_MAX, ordmax_x=INT_MIN, ordmin_y, ordmax_y}
//   sums[net*8..] = 0
// ---------------------------------------------------------------------------
__global__ void wawl_init(int* __restrict__ minmax, float* __restrict__ sums,
                          float* __restrict__ out, int N) {
  int stride = gridDim.x * blockDim.x;
  for (int n = blockIdx.x * blockDim.x + threadIdx.x; n < N; n += stride) {
    int4 mm; mm.x = INT_MAX; mm.y = INT_MIN; mm.z = INT_MAX; mm.w = INT_MIN;
    ((int4*)minmax)[n] = mm;
    float4 z; z.x = 0.f; z.y = 0.f; z.z = 0.f; z.w = 0.f;
    ((float4*)sums)[2 * n]     = z;
    ((float4*)sums)[2 * n + 1] = z;
  }
  if (blockIdx.x == 0 && threadIdx.x == 0) out[0] = 0.f;
}

// ---------------------------------------------------------------------------
// K1: per-pin ordered-int atomic min/max into L2-resident scratch.
// ---------------------------------------------------------------------------
__global__ void wawl_minmax(const float* __restrict__ pos,
                            const int* __restrict__ pin2net,
                            int* __restrict__ minmax, int P) {
  int i = blockIdx.x * blockDim.x + threadIdx.x;
  if (i >= P) return;
  float x = pos[i];
  float y = pos[P + i];
  int net = pin2net[i];
  int* mm = minmax + (size_t)net * 4;
  atomicMin(mm + 0, f2ord(x));
  atomicMax(mm + 1, f2ord(x));
  atomicMin(mm + 2, f2ord(y));
  atomicMax(mm + 3, f2ord(y));
}

// ---------------------------------------------------------------------------
// K2: TDM-staged chunk -> exp terms -> 8x global_atomic_add_f32 per pin.
//   sums[net*8 + {0..7}] = {sum_ex_x, sum_xex_x, sum_enx_x, sum_xenx_x,
//                           sum_ey_y, sum_yey_y, sum_eny_y, sum_yeny_y}
// ---------------------------------------------------------------------------
__global__ void wawl_sums(const float* __restrict__ pos,
                          const int* __restrict__ pin2net,
                          const int* __restrict__ minmax,
                          const float* __restrict__ inv_gamma,
                          float* __restrict__ sums, int P) {
  __shared__ float s_xy[2 * CHUNK];  // row 0: x chunk, row 1: y chunk
  __shared__ int   s_net[CHUNK];

  const int start = blockIdx.x * CHUNK;
  const int count = min(CHUNK, P - start);

  if (threadIdx.x < 32) {  // wave 0 issues the DMA (TDM ignores EXEC)
    uint32_t xy_off  = (uint32_t)(uintptr_t)(void*)s_xy;
    uint32_t net_off = (uint32_t)(uintptr_t)(void*)s_net;
    // pos tile: 2 rows (x at start, y at start+P), row stride = P elements
    tdm_load_4b(xy_off, pos + start, CHUNK, 2, (uint32_t)(P - start), 2u,
                (uint32_t)P);
    // pin2net tile: 1 row
    tdm_load_4b(net_off, pin2net + start, CHUNK, 1, (uint32_t)(P - start), 1u,
                (uint32_t)P);
    __builtin_amdgcn_s_wait_tensorcnt(0);
  }
  __syncthreads();

  const float ig = inv_gamma[0];
  for (int j = (int)threadIdx.x; j < count; j += THREADS) {
    float x = s_xy[j];
    float y = s_xy[CHUNK + j];
    int net = s_net[j];
    int4 mm = ((const int4*)minmax)[net];  // one b128 gather for 4 extrema
    float ex  = __expf((x - ord2f(mm.y)) * ig);
    float enx = __expf((ord2f(mm.x) - x) * ig);
    float ey  = __expf((y - ord2f(mm.w)) * ig);
    float eny = __expf((ord2f(mm.z) - y) * ig);
    float* s = sums + (size_t)net * 8;
    atomicAdd(s + 0, ex);
    atomicAdd(s + 1, x * ex);
    atomicAdd(s + 2, enx);
    atomicAdd(s + 3, x * enx);
    atomicAdd(s + 4, ey);
    atomicAdd(s + 5, y * ey);
    atomicAdd(s + 6, eny);
    atomicAdd(s + 7, y * eny);
  }
}

// ---------------------------------------------------------------------------
// K3: per-net wirelength, weighted; LDS tree reduce; one atomicAdd per block.
// ---------------------------------------------------------------------------
__global__ void wawl_finalize(const float* __restrict__ sums,
                              const float* __restrict__ net_weights,
                              const unsigned char* __restrict__ net_mask,
                              float* __restrict__ out, int N) {
  __shared__ float red[THREADS];
  int n = blockIdx.x * blockDim.x + threadIdx.x;
  float acc = 0.f;
  if (n < N) {
    float4 a = ((const float4*)sums)[2 * n];      // x: ex, xex, enx, xenx
    float4 b = ((const float4*)sums)[2 * n + 1];  // y: ey, yey, eny, yeny
    float w = net_mask[n] ? net_weights[n] : 0.f;
    float wlx = (a.x > 0.f) ? (a.y / a.x - a.w / a.z) : 0.f;
    float wly = (b.x > 0.f) ? (b.y / b.x - b.w / b.z) : 0.f;
    acc = (wlx + wly) * w;
  }
  red[threadIdx.x] = acc;
  __syncthreads();
  for (int s = THREADS / 2; s > 0; s >>= 1) {
    if ((int)threadIdx.x < s) red[threadIdx.x] += red[threadIdx.x + s];
    __syncthreads();
  }
  if (threadIdx.x == 0) atomicAdd(out, red[0]);
}

// ---------------------------------------------------------------------------
extern "C" void kernel_launch(void* const* d_in, const int* in_sizes, int n_in,
                              void* d_out, int out_size, void* d_ws,
                              size_t ws_size, hipStream_t stream) {
  const float*         pos      = (const float*)d_in[0];
  const int*           pin2net  = (const int*)d_in[1];
  const float*         weights  = (const float*)d_in[2];
  const unsigned char* net_mask = (const unsigned char*)d_in[3];
  // d_in[4] = pin_mask (unused by the reference math)
  const float*         inv_g    = (const float*)d_in[5];

  const int P = in_sizes[0] / 2;  // pos is concat(x, y)
  const int N = in_sizes[2];

  // Scratch layout (120MB for N=2.5M; L2-resident on MI455X's 192MB L2):
  //   [0, 16N)        int4 per net: {min_x, max_x, min_y, max_y} (ordered ints)
  //   [16N, 16N+32N)  8 floats per net: exp-sum accumulators
  char*  ws     = (char*)d_ws;
  int*   minmax = (int*)ws;
  float* sums   = (float*)(ws + (size_t)N * 16);
  float* out    = (float*)d_out;

  const int blkN   = (N + THREADS - 1) / THREADS;
  const int blkP   = (P + THREADS - 1) / THREADS;
  const int blkChk = (P + CHUNK - 1) / CHUNK;

  wawl_init<<<blkN, THREADS, 0, stream>>>(minmax, sums, out, N);
  wawl_minmax<<<blkP, THREADS, 0, stream>>>(pos, pin2net, minmax, P);
  wawl_sums<<<blkChk, THREADS, 0, stream>>>(pos, pin2net, minmax, inv_g, sums, P);
  wawl_finalize<<<blkN, THREADS, 0, stream>>>(sums, weights, net_mask, out, N);
}